// DiscreteKeyValueBottleneck_78580721647694
// MI455X (gfx1250) — compile-verified
//
#include <hip/hip_runtime.h>

#define H_  12
#define C_  4096
#define DK_ 64
#define DV_ 64
#define N_  4096          // B*T
#define OUTSTRIDE 768     // H*DK = H*DV
#define MTILE 128
#define CTILE 128
#define NCHUNK (C_ / CTILE)
#define ESTRIDE 68        // 64 + 4 padding -> conflict-free LDS reads

typedef __attribute__((ext_vector_type(2))) float v2f;
typedef __attribute__((ext_vector_type(8))) float v8f;
typedef unsigned long long u64;

__global__ __launch_bounds__(256) void dkvb_e2_kernel(const float* __restrict__ ke,
                                                      float* __restrict__ e2) {
    int i = blockIdx.x * blockDim.x + threadIdx.x;
    if (i >= H_ * C_) return;
    const float4* p = reinterpret_cast<const float4*>(ke + (size_t)i * DK_);
    float s = 0.f;
#pragma unroll
    for (int j = 0; j < DK_ / 4; ++j) {
        float4 v = p[j];
        s += v.x * v.x + v.y * v.y + v.z * v.z + v.w * v.w;
    }
    e2[i] = s;
}

// order-preserving float -> u32 map (finite inputs): a<b  <=>  map(a)<map(b)
__device__ __forceinline__ unsigned sortable(float f) {
    unsigned b = __float_as_uint(f);
    return b ^ (unsigned)(((int)b >> 31) | 0x80000000);
}

__global__ __launch_bounds__(256) void dkvb_main_kernel(
    const float* __restrict__ x,          // [N, 768]
    const float* __restrict__ key_embed,  // [H, C, DK]
    const float* __restrict__ values,     // [H, C, DV]
    const float* __restrict__ e2g,        // [H, C]
    float* __restrict__ out)              // [N, 768]
{
    __shared__ float Es[CTILE * ESTRIDE];
    __shared__ float e2s[CTILE];
    __shared__ int   idx_s[MTILE];

    const int h    = blockIdx.y;
    const int n0   = blockIdx.x * MTILE;
    const int tid  = threadIdx.x;
    const int wave = tid >> 5;
    const int lane = tid & 31;
    const int hi   = lane >> 4;   // half-wave id
    const int lo16 = lane & 15;
    const int kb   = hi * 2;      // K sub-offset for this half (f32 16x16x4 layout)

    // ---- A fragments: wave's 16 rows of X, K=64, kept in registers ----
    const int m = 16 * wave + lo16;
    const float* xg = x + (size_t)(n0 + m) * OUTSTRIDE + h * DK_ + kb;
    v2f a_frag[16];
#pragma unroll
    for (int s = 0; s < 16; ++s)
        a_frag[s] = *reinterpret_cast<const v2f*>(xg + 4 * s);

    // per-row running best over this lane's column subset (ascending c order,
    // so strict < gives first-occurrence tie-breaking within a lane)
    float best_val[8];
    int   best_idx[8];
#pragma unroll
    for (int r = 0; r < 8; ++r) { best_val[r] = __builtin_inff(); best_idx[r] = 0; }

    const float*  keh  = key_embed + (size_t)h * C_ * DK_;
    const float4* esrc = reinterpret_cast<const float4*>(keh);
    const int     q0   = tid;               // this thread's first float4 slot
    const int     row0 = q0 >> 4, col40 = q0 & 15;

    // prologue: fetch chunk 0 into registers
    float4 stg[8];
#pragma unroll
    for (int i = 0; i < 8; ++i) stg[i] = esrc[i * 256 + q0];
    float e2r = (tid < CTILE) ? e2g[h * C_ + tid] : 0.f;

    for (int cc = 0; cc < NCHUNK; ++cc) {
        const int c0 = cc * CTILE;
        // commit staged chunk to LDS
#pragma unroll
        for (int i = 0; i < 8; ++i)
            *reinterpret_cast<float4*>(&Es[(row0 + i * 16) * ESTRIDE + col40 * 4]) = stg[i];
        if (tid < CTILE) e2s[tid] = e2r;
        __syncthreads();

        // kick off next chunk's global loads (hide under WMMA work)
        if (cc + 1 < NCHUNK) {
            const float4* nsrc = esrc + (size_t)(c0 + CTILE) * (DK_ / 4);
#pragma unroll
            for (int i = 0; i < 8; ++i) stg[i] = nsrc[i * 256 + q0];
            e2r = (tid < CTILE) ? e2g[h * C_ + c0 + CTILE + tid] : 0.f;
        }

#pragma unroll 1
        for (int ct = 0; ct < 8; ++ct) {
            // B fragments from LDS: B[k,n] = E[c0+ct*16+n, k]
            const float* ep = &Es[(ct * 16 + lo16) * ESTRIDE + kb];
            v2f b_frag[16];
#pragma unroll
            for (int s = 0; s < 16; ++s)
                b_frag[s] = *reinterpret_cast<const v2f*>(ep + 4 * s);

            v8f d = {0.f, 0.f, 0.f, 0.f, 0.f, 0.f, 0.f, 0.f};
#pragma unroll
            for (int s = 0; s < 16; ++s) {
                d = __builtin_amdgcn_wmma_f32_16x16x4_f32(
                        false, a_frag[s], false, b_frag[s],
                        (short)0, d, false, false);
            }

            // branchless per-lane running argmin (float cmp + selects only)
            const int   c    = c0 + ct * 16 + lo16;
            const float ecol = e2s[ct * 16 + lo16];
#pragma unroll
            for (int r = 0; r < 8; ++r) {
                float val  = __builtin_fmaf(-2.0f, d[r], ecol); // row m = 16w + r + 8*hi
                bool  take = val < best_val[r];
                best_val[r] = take ? val : best_val[r];
                best_idx[r] = take ? c   : best_idx[r];
            }
        }
        __syncthreads();
    }

    // epilogue: pack (score, idx) once, butterfly min across the 16 lanes
    // holding each row; lowest index wins exact ties.
#pragma unroll
    for (int r = 0; r < 8; ++r) {
        u64 k = ((u64)sortable(best_val[r]) << 32) | (unsigned)best_idx[r];
#pragma unroll
        for (int off = 1; off < 16; off <<= 1) {
            unsigned klo = (unsigned)k, khi = (unsigned)(k >> 32);
            unsigned olo = __shfl_xor(klo, off, 32);
            unsigned ohi = __shfl_xor(khi, off, 32);
            u64 o = ((u64)ohi << 32) | olo;
            k = o < k ? o : k;
        }
        best_idx[r] = (int)(unsigned)k;
    }
    if (lo16 == 0) {
#pragma unroll
        for (int r = 0; r < 8; ++r)
            idx_s[16 * wave + r + 8 * hi] = best_idx[r];
    }
    __syncthreads();

    // gather values[h, ind, :] -> out rows (2 threads per row, float4)
    {
        const int row  = tid >> 1;
        const int half = tid & 1;
        const int ind  = idx_s[row];
        const float4* src = reinterpret_cast<const float4*>(
            values + ((size_t)h * C_ + ind) * DV_ + half * 32);
        float4* dst = reinterpret_cast<float4*>(
            out + (size_t)(n0 + row) * OUTSTRIDE + h * DV_ + half * 32);
#pragma unroll
        for (int i = 0; i < 8; ++i) dst[i] = src[i];
    }
}

extern "C" void kernel_launch(void* const* d_in, const int* in_sizes, int n_in,
                              void* d_out, int out_size, void* d_ws, size_t ws_size,
                              hipStream_t stream) {
    const float* x         = (const float*)d_in[0];
    // d_in[1] = mask (unused by reference), d_in[4] = key_optim (unused)
    const float* key_embed = (const float*)d_in[2];
    const float* values    = (const float*)d_in[3];
    float*       out       = (float*)d_out;
    float*       e2        = (float*)d_ws;   // H_*C_ floats = 192 KiB scratch

    dkvb_e2_kernel<<<(H_ * C_ + 255) / 256, 256, 0, stream>>>(key_embed, e2);

    dim3 grid(N_ / MTILE, H_);
    dkvb_main_kernel<<<grid, 256, 0, stream>>>(x, key_embed, values, e2, out);
}